// Consistent_loss_up_25288767439314
// MI455X (gfx1250) — compile-verified
//
#include <hip/hip_runtime.h>
#include <hip/hip_bf16.h>

#define HH 256
#define WW 256
#define TILE 64            // columns (threads) per block
#define BSTRIDE 53         // 51 bins padded to odd stride -> LDS conflict friendly
#define MIN_DISP 0.0235f
#define THRESH 0.2f

typedef __attribute__((ext_vector_type(2))) float v2f;
typedef __attribute__((ext_vector_type(8))) float v8f;

// One block = one (batch, 64-column tile). Lane t owns column x0+t.
// Phase 1: stream up[b, :, x0..x0+63] (coalesced), scatter-max into private LDS
//          bins using the monotonicity trick (first write wins, no LDS RMW).
// Phase 2: stream left/right rows (coalesced), masked-L1 accumulate.
__global__ __launch_bounds__(TILE) void splat_loss_kernel(
    const float* __restrict__ up, const float* __restrict__ left,
    const float* __restrict__ right, float* __restrict__ partial)
{
#pragma clang fp contract(off)
    __shared__ float binsL[TILE * BSTRIDE];   // ~13.2 KB
    __shared__ float binsR[TILE * BSTRIDE];   // ~13.2 KB
    __shared__ float vtab[HH];                // splat value per row j
    __shared__ float red[TILE];

    const int tid = threadIdx.x;
    const int b   = blockIdx.x >> 2;
    const int x0  = (blockIdx.x & 3) * TILE;

    // Zero bins (strided, whole array) and build the value table.
    for (int k = tid; k < TILE * BSTRIDE; k += TILE) {
        binsL[k] = 0.0f;
        binsR[k] = 0.0f;
    }
    for (int j = tid; j < HH; j += TILE) {
        vtab[j] = (j <= 128) ? ((128.0f - (float)j) / 60.0f)
                             : (((float)j - 128.0f) / 60.0f);
    }
    __syncthreads();

    // ---------------- Phase 1: scatter-max splat ----------------
    const float* upcol = up + ((size_t)b * (HH * WW)) + (size_t)x0 + tid;
    float* myL = &binsL[tid * BSTRIDE];
    float* myR = &binsR[tid * BSTRIDE];

    // Left: j = 0..128, value (128-j)/60 strictly decreasing -> first hit wins.
    {
        unsigned long long wr = 0ull;
        for (int j = 0; j <= 128; ++j) {
            if (j < 112) __builtin_prefetch(upcol + (size_t)(j + 16) * WW, 0, 0);
            float u = upcol[(size_t)j * WW];
            if (u >= MIN_DISP) {
                float f = u * 50.0f + 110.0f;      // contract(off): mul then add, like XLA
                int bin = (int)rintf(f) - 110;     // round-half-even == jnp.round
                if ((unsigned)bin <= 50u && !((wr >> bin) & 1ull)) {
                    wr |= 1ull << bin;
                    myL[bin] = vtab[j];
                }
            }
        }
    }
    // Right: j = 255..129, value (j-128)/60 decreasing as j falls -> first hit wins.
    {
        unsigned long long wr = 0ull;
        for (int j = 255; j >= 129; --j) {
            if (j >= 145) __builtin_prefetch(upcol + (size_t)(j - 16) * WW, 0, 0);
            float u = upcol[(size_t)j * WW];
            if (u >= MIN_DISP) {
                float f = u * 50.0f + 110.0f;
                int bin = (int)rintf(f) - 110;
                if ((unsigned)bin <= 50u && !((wr >> bin) & 1ull)) {
                    wr |= 1ull << bin;
                    myR[bin] = vtab[j];
                }
            }
        }
    }
    __syncthreads();

    // ---------------- Phase 2: masked L1 (coalesced global reads) ----------------
    float acc = 0.0f;
    const float* lb = left  + (size_t)b * (HH * WW);
    const float* rb = right + (size_t)b * (HH * WW);
    for (int xx = 0; xx < TILE; ++xx) {
        const float* lrow = lb + (size_t)(x0 + xx) * WW;
        const float* rrow = rb + (size_t)(x0 + xx) * WW;
        const float* bL = &binsL[xx * BSTRIDE];
        const float* bR = &binsR[xx * BSTRIDE];
#pragma unroll
        for (int k = 0; k < 4; ++k) {
            int y = k * TILE + tid;                 // 64 lanes sweep the row: coalesced
            unsigned bi = (unsigned)(y - 110);
            float vL = (bi <= 50u) ? bL[bi] : 0.0f;
            float vR = (bi <= 50u) ? bR[bi] : 0.0f;
            float dL = fabsf(vL - lrow[y]);
            float dR = fabsf(vR - rrow[y]);
            acc += (dL < THRESH) ? dL : 0.0f;
            acc += (dR < THRESH) ? dR : 0.0f;
        }
    }

    // Fixed-order block reduction -> one partial per block (deterministic).
    red[tid] = acc;
    __syncthreads();
    for (int s = TILE / 2; s > 0; s >>= 1) {
        if (tid < s) red[tid] += red[tid + s];
        __syncthreads();
    }
    if (tid == 0) partial[blockIdx.x] = red[0];
}

// Final reduction of block partials on the matrix pipe:
// D = A x ones + C with V_WMMA_F32_16X16X4_F32. With B = all-ones, each D row
// accumulates the sum of that row of A; chaining D->C across chunks gives a
// deterministic, hardware-ordered sum of all partials. Lane 0 holds rows 0..7
// (in its 8 acc VGPRs), lane 16 holds rows 8..15.
__global__ __launch_bounds__(32) void reduce_partials_kernel(
    const float* __restrict__ partial, int n, float scale, float* __restrict__ out)
{
    const int lane = threadIdx.x;
    v8f c = {0.0f, 0.0f, 0.0f, 0.0f, 0.0f, 0.0f, 0.0f, 0.0f};
    v2f onev; onev.x = 1.0f; onev.y = 1.0f;

    const int chunks = (n + 63) >> 6;              // 64 floats per WMMA (2 VGPRs x 32 lanes)
    for (int ch = 0; ch < chunks; ++ch) {
        int i0 = (ch << 6) + lane;
        int i1 = i0 + 32;
        v2f a;
        a.x = (i0 < n) ? partial[i0] : 0.0f;       // cndmask select, wave stays converged
        a.y = (i1 < n) ? partial[i1] : 0.0f;
#if __has_builtin(__builtin_amdgcn_wmma_f32_16x16x4_f32)
        c = __builtin_amdgcn_wmma_f32_16x16x4_f32(
                /*neg_a=*/false, a, /*neg_b=*/false, onev,
                /*c_mod=*/(short)0, c, /*reuse_a=*/false, /*reuse_b=*/false);
#else
        c[0] += a.x + a.y;
#endif
    }

    float t = c[0] + c[1] + c[2] + c[3] + c[4] + c[5] + c[6] + c[7];
#if __has_builtin(__builtin_amdgcn_wmma_f32_16x16x4_f32)
    float s = __shfl(t, 0, 32) + __shfl(t, 16, 32);  // rows 0..7 + rows 8..15
#else
    for (int off = 16; off > 0; off >>= 1) t += __shfl_down(t, off, 32);
    float s = __shfl(t, 0, 32);
#endif
    if (lane == 0) out[0] = s * scale;               // N = 2^25 -> exact scaling
}

extern "C" void kernel_launch(void* const* d_in, const int* in_sizes, int n_in,
                              void* d_out, int out_size, void* d_ws, size_t ws_size,
                              hipStream_t stream)
{
    const float* up    = (const float*)d_in[0];
    const float* left  = (const float*)d_in[1];
    const float* right = (const float*)d_in[2];
    float* out     = (float*)d_out;
    float* partial = (float*)d_ws;

    const int total   = in_sizes[0];           // B * 1 * H * W
    const int B       = total / (HH * WW);
    const int nblocks = B * 4;                 // 4 column tiles per batch image

    splat_loss_kernel<<<nblocks, TILE, 0, stream>>>(up, left, right, partial);

    const float scale = 1.0f / (float)total;   // (sumL + sumR) / N == meanL + meanR
    reduce_partials_kernel<<<1, 32, 0, stream>>>(partial, nblocks, scale, out);
}